// GraphConvolution_51290499448997
// MI455X (gfx1250) — compile-verified
//
#include <hip/hip_runtime.h>
#include <math.h>

#define N_NODES 100000
#define N_EDGES 1600000
#define D_FEAT  256
#define UNITS   128

typedef __attribute__((ext_vector_type(2))) float v2f;
typedef __attribute__((ext_vector_type(8))) float v8f;

// ---------------------------------------------------------------- zero / relu
__global__ __launch_bounds__(256) void zero_kernel(float4* __restrict__ out, int n4) {
    int i = blockIdx.x * 256 + threadIdx.x;
    if (i < n4) out[i] = make_float4(0.f, 0.f, 0.f, 0.f);
}

__global__ __launch_bounds__(256) void relu_kernel(float4* __restrict__ out, int n4) {
    int i = blockIdx.x * 256 + threadIdx.x;
    if (i < n4) {
        float4 v = out[i];
        v.x = fmaxf(v.x, 0.f); v.y = fmaxf(v.y, 0.f);
        v.z = fmaxf(v.z, 0.f); v.w = fmaxf(v.w, 0.f);
        out[i] = v;
    }
}

// ---------------------------------------------------------------- gate GEMV
// one wave32 per row: g[row] = sigmoid(dot(x[row,:], wg) + bg)
__global__ __launch_bounds__(256) void gate_kernel(const float* __restrict__ x,
                                                   const float* __restrict__ wg,
                                                   const float* __restrict__ bg,
                                                   float* __restrict__ g) {
    const int lane = threadIdx.x & 31;
    const int wave = threadIdx.x >> 5;
    const int row  = blockIdx.x * 8 + wave;            // N_NODES % 8 == 0
    const float* xr = x + (size_t)row * D_FEAT;
    float s = 0.f;
#pragma unroll
    for (int j = 0; j < D_FEAT; j += 32)
        s = fmaf(xr[j + lane], wg[j + lane], s);
#pragma unroll
    for (int off = 16; off > 0; off >>= 1)
        s += __shfl_xor(s, off, 32);
    if (lane == 0)
        g[row] = 1.0f / (1.0f + __expf(-(s + bg[0])));
}

// ---------------------------------------------------------------- FP32 WMMA GEMM
// block = 8 waves; block computes rows [16b,16b+16) x all 128 units.
// wave w computes the 16x16 tile at columns [16w,16w+16) with V_WMMA_F32_16X16X4_F32,
// iterating K=256 in steps of 4. Epilogue fuses bias add and gate scaling.
__global__ __launch_bounds__(256) void gemm_gate_kernel(const float* __restrict__ x,
                                                        const float* __restrict__ W,    // [D_FEAT, UNITS]
                                                        const float* __restrict__ bias, // [UNITS]
                                                        const float* __restrict__ g,    // [N_NODES]
                                                        float* __restrict__ h) {        // [N_NODES, UNITS]
    const int lane = threadIdx.x & 31;
    const int wave = threadIdx.x >> 5;
    const int m16  = lane & 15;
    const int half = lane >> 4;                        // 0: lanes 0-15, 1: lanes 16-31
    const size_t row0 = (size_t)blockIdx.x * 16;       // N_NODES % 16 == 0 -> EXEC all-1s
    const float* xr = x + (row0 + (size_t)m16) * D_FEAT;
    const int ncol = wave * 16 + m16;

    v8f c = {};
#pragma unroll 4
    for (int k = 0; k < D_FEAT; k += 4) {
        const int ka = k + 2 * half;                   // A: VGPR v holds K = 2*half + v
        v2f a;
        a.x = xr[ka];
        a.y = xr[ka + 1];
        v2f b;                                         // B: VGPR v holds K = 2*half + v, N = lane%16
        b.x = W[(size_t)ka * UNITS + ncol];
        b.y = W[(size_t)(ka + 1) * UNITS + ncol];
        c = __builtin_amdgcn_wmma_f32_16x16x4_f32(false, a, false, b, (short)0, c, false, false);
    }

    const float bc = bias[ncol];
#pragma unroll
    for (int r = 0; r < 8; ++r) {                      // C: VGPR r -> M = r + 8*half, N = lane%16
        const size_t row = row0 + (size_t)(r + 8 * half);
        h[row * UNITS + ncol] = g[row] * (c[r] + bc);
    }
}

// ---------------------------------------------------------------- SpMM scatter
// one wave32 per edge; lane l handles units [4l, 4l+4): float4 gather of h[col],
// scale by edge value, hardware f32 atomic-add into out[row].
__global__ __launch_bounds__(256) void spmm_kernel(const int*   __restrict__ rows,
                                                   const int*   __restrict__ cols,
                                                   const float* __restrict__ vals,
                                                   const float* __restrict__ h,
                                                   float*       __restrict__ out) {
    const int lane = threadIdx.x & 31;
    const int e = (blockIdx.x * 256 + (int)threadIdx.x) >> 5;   // grid sized exactly
    const int r = rows[e];
    const int c = cols[e];
    const float v = vals[e];
    const float4 hv = ((const float4*)(h + (size_t)c * UNITS))[lane];
    float* op = out + (size_t)r * UNITS + (size_t)lane * 4;
    unsafeAtomicAdd(op + 0, v * hv.x);
    unsafeAtomicAdd(op + 1, v * hv.y);
    unsafeAtomicAdd(op + 2, v * hv.z);
    unsafeAtomicAdd(op + 3, v * hv.w);
}

// ---------------------------------------------------------------- launch
extern "C" void kernel_launch(void* const* d_in, const int* in_sizes, int n_in,
                              void* d_out, int out_size, void* d_ws, size_t ws_size,
                              hipStream_t stream) {
    const float* x  = (const float*)d_in[0];
    const float* ev = (const float*)d_in[1];   // [2, E]
    const float* W  = (const float*)d_in[2];   // [2, 256, 128]
    const float* b  = (const float*)d_in[3];   // [2, 128]
    const float* Wg = (const float*)d_in[4];   // [2, 256, 1]
    const float* bg = (const float*)d_in[5];   // [2, 1]
    const int*   er = (const int*)d_in[6];     // [2, E]
    const int*   ec = (const int*)d_in[7];     // [2, E]
    float* out = (float*)d_out;

    float* h = (float*)d_ws;                          // [N_NODES, UNITS] = 51.2 MB
    float* g = h + (size_t)N_NODES * UNITS;           // [N_NODES]        = 0.4 MB

    const int n4 = N_NODES * UNITS / 4;
    zero_kernel<<<(n4 + 255) / 256, 256, 0, stream>>>((float4*)out, n4);

    for (int i = 0; i < 2; ++i) {
        gate_kernel<<<N_NODES / 8, 256, 0, stream>>>(
            x, Wg + (size_t)i * D_FEAT, bg + i, g);
        gemm_gate_kernel<<<N_NODES / 16, 256, 0, stream>>>(
            x, W + (size_t)i * D_FEAT * UNITS, b + (size_t)i * UNITS, g, h);
        spmm_kernel<<<N_EDGES / 8, 256, 0, stream>>>(
            er + (size_t)i * N_EDGES, ec + (size_t)i * N_EDGES,
            ev + (size_t)i * N_EDGES, h, out);
    }

    relu_kernel<<<(n4 + 255) / 256, 256, 0, stream>>>((float4*)out, n4);
}